// Probability_layer_gaussian_23837068493460
// MI455X (gfx1250) — compile-verified
//
#include <hip/hip_runtime.h>

// ---------------------------------------------------------------------------
// Fused dual-head Gaussian layer for MI455X (gfx1250, wave32, WMMA + TDM).
//   mean    = x @ W_mu + b_mu
//   log_std = x @ W_ls + b_ls
//   var     = exp(2*log_std)
//   z       = mean + exp(log_std) * eps   (eps: deterministic hash Box-Muller)
//
// HBM-bound (~470 MB @ 23.3 TB/s => ~20us floor). Strategy:
//  * fp32 WMMA (V_WMMA_F32_16X16X4_F32): compute is ~nothing vs memory, so
//    keep full reference precision.
//  * Tensor Data Mover (tensor_load_to_lds) streams the x-tile and both
//    weight tiles straight into LDS (no VGPR staging, TENSORcnt-tracked),
//    double-buffered so stage s+1 DMA overlaps stage s WMMAs.
//  * A-tile LDS row stride padded to 20 floats via the TDM pad feature
//    (pad_interval=16 DW, pad_amount=4 DW): bank-conflict-free frag reads.
// ---------------------------------------------------------------------------

typedef __attribute__((ext_vector_type(2))) float v2f;
typedef __attribute__((ext_vector_type(8))) float v8f;
typedef __attribute__((ext_vector_type(4))) unsigned int u32x4;
typedef __attribute__((ext_vector_type(8))) int i32x8;
typedef __attribute__((ext_vector_type(4))) int i32x4;

#define NROWS 65536
#define DIN   1024
#define DOUT  256
#define BM    64
#define BN    64
#define BK    16
#define APAD  20                      // A-tile row stride in floats (16 data + 4 pad)
#define ABYTES   (BM * APAD * 4)      // 5120
#define BBYTES   (BK * BN * 4)        // 4096
#define BUFFLOATS (BM * APAD + 2 * BK * BN)   // 3328 floats = 13312 B per buffer

#if __has_builtin(__builtin_amdgcn_tensor_load_to_lds)
#define HAVE_TDM 1
#else
#define HAVE_TDM 0
#endif

__device__ __forceinline__ unsigned h32(unsigned x) {
  x ^= x >> 17; x *= 0xED5AD4BBu;
  x ^= x >> 11; x *= 0xAC4C1B51u;
  x ^= x >> 15; x *= 0x31848BABu;
  x ^= x >> 14;
  return x;
}

__device__ __forceinline__ float gauss_eps(unsigned idx) {
  unsigned r1 = h32(idx * 0x9E3779B9u + 0x7F4A7C15u);
  unsigned r2 = h32(idx ^ 0x85EBCA6Bu);
  float u1 = (float)(r1 >> 8) * (1.0f / 16777216.0f) + (0.5f / 16777216.0f);
  float u2 = (float)(r2 >> 8) * (1.0f / 16777216.0f);
  float r  = __builtin_sqrtf(-2.0f * __logf(u1));
  return r * __cosf(6.2831853071795864f * u2);
}

#if HAVE_TDM
// Build D# group1 for a 2D fp32 tile load (data_size=4B code 2).
__device__ __forceinline__ i32x8 make_g1(int flags, unsigned td0, unsigned td1,
                                         unsigned tile0, unsigned tile1,
                                         unsigned stride0) {
  i32x8 g1;
  g1[0] = flags;                                            // mask/ds/pad bits
  g1[1] = (int)((td0 & 0xFFFFu) << 16);                     // barrier_addr=0 | td0.lo
  g1[2] = (int)(((td0 >> 16) & 0xFFFFu) | ((td1 & 0xFFFFu) << 16));
  g1[3] = (int)(((td1 >> 16) & 0xFFFFu) | (tile0 << 16));   // td1.hi | tile_dim0
  g1[4] = (int)(tile1 & 0xFFFFu);                           // tile_dim1 | tile_dim2=0
  g1[5] = (int)stride0;                                     // dim0 stride lo32
  g1[6] = 0;                                                // stride hi | d1stride lo
  g1[7] = 0;
  return g1;
}

// Issue one TDM 2D tile load: group0 carries LDS byte offset + global address.
// This toolchain exposes the 6-arg builtin:
//   (u32x4 g0, i32x8 g1, i32x4 g2, i32x4 g3, i32x8 extra, i32 cpol)
__device__ __forceinline__ void tdm_load_2d(unsigned lds_off, const void* gptr,
                                            i32x8 g1) {
  unsigned long long ga = (unsigned long long)(uintptr_t)gptr;
  u32x4 g0;
  g0.x = 1u;                                           // count=1, user descriptor
  g0.y = lds_off;                                      // lds_addr (bytes)
  g0.z = (unsigned)ga;                                 // global_addr[31:0]
  g0.w = (unsigned)((ga >> 32) & 0x1FFFFFFull) | (2u << 30);  // addr[56:32]|type=2
  i32x4 z4 = {0, 0, 0, 0};
  i32x8 z8 = {0, 0, 0, 0, 0, 0, 0, 0};
  __builtin_amdgcn_tensor_load_to_lds(g0, g1, z4, z4, z8, 0);
}

// data_size=4B(2); A adds pad: enable<<20, interval 16DW(3)<<22, amount 4DW(3)<<25
#define A_FLAGS ((2 << 16) | (1 << 20) | (3 << 22) | (3 << 25))
#define B_FLAGS (2 << 16)
#endif  // HAVE_TDM

__global__ __launch_bounds__(256) void gaussian_heads_wmma(
    const float* __restrict__ x,
    const float* __restrict__ Wmu, const float* __restrict__ bmu,
    const float* __restrict__ Wls, const float* __restrict__ bls,
    float* __restrict__ out)
{
  __shared__ float lds[2 * BUFFLOATS];   // double-buffered {A | Bmu | Bls}

  const int tid  = threadIdx.x;
  const int lane = tid & 31;             // wave32
  const int wave = tid >> 5;             // 8 waves

  const int bm0 = blockIdx.x * BM;       // grid.x = 1024
  const int bn0 = blockIdx.y * BN;       // grid.y = 4

  // Wave -> tiles: 1 M-tile, 2 N-tiles, both heads => 4 accumulators.
  const int mtile = wave & 3;
  const int nt0   = (wave >> 2) * 2;

  // Fragment lane mapping (ISA 32-bit 16x4 A layout; B symmetric in N):
  const int frow = lane & 15;
  const int kq   = (lane >> 4) << 1;     // 0 or 2

  v8f accm0 = {}, accm1 = {}, accl0 = {}, accl1 = {};

  const int NSTAGE = DIN / BK;           // 64

#if HAVE_TDM
  const unsigned lds_base = (unsigned)(uintptr_t)&lds[0];  // flat lo32 == LDS offset
  const i32x8 g1a = make_g1(A_FLAGS, DIN, NROWS, BK, BM, DIN);
  const i32x8 g1b = make_g1(B_FLAGS, DOUT, DIN, BN, BK, DOUT);

  // Scalar-branch gate: tensor ops ignore EXEC, so gate with a uniform SGPR.
  const int wv = __builtin_amdgcn_readfirstlane(wave);

  // Prime buffer 0.
  if (wv == 0) {
    const unsigned off = lds_base;
    tdm_load_2d(off,                    x   + (size_t)bm0 * DIN,        g1a);
    tdm_load_2d(off + ABYTES,           Wmu + (size_t)bn0,              g1b);
    tdm_load_2d(off + ABYTES + BBYTES,  Wls + (size_t)bn0,              g1b);
  }
  __builtin_amdgcn_s_wait_tensorcnt(0);
  __syncthreads();
#else
  const int a_row = tid >> 2, a_col = (tid & 3) << 2;
  const int b_row = tid >> 4, b_col = (tid & 15) << 2;
  const float* xg  = x   + (size_t)(bm0 + a_row) * DIN + a_col;
  const float* wmg = Wmu + (size_t)b_row * DOUT + bn0 + b_col;
  const float* wlg = Wls + (size_t)b_row * DOUT + bn0 + b_col;
#endif

  for (int s = 0; s < NSTAGE; ++s) {
#if HAVE_TDM
    const int buf = s & 1;
    // Kick off stage s+1 DMA into the other buffer; overlaps the WMMAs below.
    if (wv == 0 && (s + 1) < NSTAGE) {
      const int k1 = (s + 1) * BK;
      const unsigned off = lds_base + (unsigned)((buf ^ 1) * BUFFLOATS * 4);
      tdm_load_2d(off,                   x   + (size_t)bm0 * DIN + k1,        g1a);
      tdm_load_2d(off + ABYTES,          Wmu + (size_t)k1 * DOUT + bn0,       g1b);
      tdm_load_2d(off + ABYTES + BBYTES, Wls + (size_t)k1 * DOUT + bn0,       g1b);
    }
    const float* As = &lds[buf * BUFFLOATS];
    const float* Bm = As + BM * APAD;
    const float* Bl = Bm + BK * BN;
#else
    const int k0 = s * BK;
    float* As = &lds[0];
    float* Bm = As + BM * APAD;
    float* Bl = Bm + BK * BN;
    *(float4*)&As[a_row * APAD + a_col] = *(const float4*)(xg + k0);
    *(float4*)&Bm[b_row * BN + b_col]   = *(const float4*)(wmg + (size_t)k0 * DOUT);
    *(float4*)&Bl[b_row * BN + b_col]   = *(const float4*)(wlg + (size_t)k0 * DOUT);
    __syncthreads();
#endif

    const float* arow = &As[(mtile * 16 + frow) * APAD];
#pragma unroll
    for (int kk = 0; kk < BK; kk += 4) {
      const int ka = kk + kq;
      v2f a; a.x = arow[ka]; a.y = arow[ka + 1];

      v2f b0; b0.x = Bm[ka * BN + nt0 * 16 + frow];
              b0.y = Bm[(ka + 1) * BN + nt0 * 16 + frow];
      accm0 = __builtin_amdgcn_wmma_f32_16x16x4_f32(
          false, a, false, b0, (short)0, accm0, false, false);

      v2f b1; b1.x = Bm[ka * BN + (nt0 + 1) * 16 + frow];
              b1.y = Bm[(ka + 1) * BN + (nt0 + 1) * 16 + frow];
      accm1 = __builtin_amdgcn_wmma_f32_16x16x4_f32(
          false, a, false, b1, (short)0, accm1, false, false);

      v2f c0; c0.x = Bl[ka * BN + nt0 * 16 + frow];
              c0.y = Bl[(ka + 1) * BN + nt0 * 16 + frow];
      accl0 = __builtin_amdgcn_wmma_f32_16x16x4_f32(
          false, a, false, c0, (short)0, accl0, false, false);

      v2f c1; c1.x = Bl[ka * BN + (nt0 + 1) * 16 + frow];
              c1.y = Bl[(ka + 1) * BN + (nt0 + 1) * 16 + frow];
      accl1 = __builtin_amdgcn_wmma_f32_16x16x4_f32(
          false, a, false, c1, (short)0, accl1, false, false);
    }

#if HAVE_TDM
    __builtin_amdgcn_s_wait_tensorcnt(0);  // next-stage tiles landed
    __syncthreads();                       // all waves done with this buffer
#else
    __syncthreads();
#endif
  }

  // ---- fused epilogue: bias + exp + sampling, straight to HBM ----
  const size_t NT = (size_t)NROWS * DOUT;
  const int mbase = bm0 + mtile * 16 + ((lane >> 4) << 3);

#pragma unroll
  for (int t = 0; t < 2; ++t) {
    v8f mu = t ? accm1 : accm0;
    v8f ls = t ? accl1 : accl0;
    const int   c     = bn0 + (nt0 + t) * 16 + frow;
    const float bmu_c = bmu[c];
    const float bls_c = bls[c];
#pragma unroll
    for (int j = 0; j < 8; ++j) {
      const int    m   = mbase + j;
      const size_t idx = (size_t)m * DOUT + c;
      float mean = mu[j] + bmu_c;
      float lstd = ls[j] + bls_c;
      float stdv = __expf(lstd);
      float var  = stdv * stdv;            // exp(2*log_std)
      float z    = __builtin_fmaf(stdv, gauss_eps((unsigned)idx), mean);
      out[idx]          = mean;
      out[NT + idx]     = var;
      out[2 * NT + idx] = z;
    }
  }
}

extern "C" void kernel_launch(void* const* d_in, const int* in_sizes, int n_in,
                              void* d_out, int out_size, void* d_ws, size_t ws_size,
                              hipStream_t stream) {
  const float* x   = (const float*)d_in[0];
  const float* Wmu = (const float*)d_in[1];
  const float* bmu = (const float*)d_in[2];
  const float* Wls = (const float*)d_in[3];
  const float* bls = (const float*)d_in[4];
  float* out = (float*)d_out;

  dim3 grid(NROWS / BM, DOUT / BN);   // (1024, 4)
  gaussian_heads_wmma<<<grid, 256, 0, stream>>>(x, Wmu, bmu, Wls, bls, out);
}